// CutoffModule_54400055771276
// MI455X (gfx1250) — compile-verified
//
#include <hip/hip_runtime.h>

// ---------------------------------------------------------------------------
// Problem constants (match reference: N=32, C=512, H=W=64, D=4, r=16)
// ---------------------------------------------------------------------------
#define N_    32
#define C_    512
#define HW_   4096          // 64*64
#define D_    4
#define CR_   32            // C / r  (MLP hidden)
#define CD_   2048          // C * D  (MLP out)
#define BLK_  128           // C / D  (top-k count)

typedef __attribute__((ext_vector_type(2))) float v2f;
typedef __attribute__((ext_vector_type(4))) float v4f;
typedef __attribute__((ext_vector_type(8))) float v8f;

// ---------------------------------------------------------------------------
// Kernel 1: avg+max pooling over H*W, one wave32 per (n,c) plane.
// 16384 planes * 16KB = 256MB streamed once; float4 loads, shuffle reduction.
// ---------------------------------------------------------------------------
__global__ void pool_kernel(const float* __restrict__ x,
                            float* __restrict__ avgp,
                            float* __restrict__ maxp) {
    const int wave = (blockIdx.x * blockDim.x + threadIdx.x) >> 5;   // (n*C + c)
    const int lane = threadIdx.x & 31;
    const v4f* b4 = (const v4f*)(x + (size_t)wave * HW_);
    float s = 0.0f;
    float m = -3.4028235e38f;
    #pragma unroll
    for (int i = 0; i < 32; ++i) {                // 32 iters * 32 lanes * 4 = 4096
        v4f v = b4[i * 32 + lane];
        s += v.x + v.y + v.z + v.w;
        m = fmaxf(m, fmaxf(fmaxf(v.x, v.y), fmaxf(v.z, v.w)));
    }
    #pragma unroll
    for (int off = 16; off; off >>= 1) {          // wave32 reduction
        s += __shfl_xor(s, off, 32);
        m = fmaxf(m, __shfl_xor(m, off, 32));
    }
    if (lane == 0) {
        avgp[wave] = s * (1.0f / HW_);
        maxp[wave] = m;
    }
}

// ---------------------------------------------------------------------------
// Kernel 2: GEMM1 via v_wmma_f32_16x16x4_f32.
// H_in = relu(P@W1 + b1) for P in {avg,max}; HS = H_avg + H_max  -> [32][32].
// One block, 8 waves: wave = (input,tile_m,tile_n). K=512 => 128 WMMA steps.
// ---------------------------------------------------------------------------
__global__ void gemm1_kernel(const float* __restrict__ avgp,
                             const float* __restrict__ maxp,
                             const float* __restrict__ W1,
                             const float* __restrict__ b1,
                             float* __restrict__ hs_out) {
    __shared__ float Hld[2][32][32];
    const int tid   = threadIdx.x;
    const int w     = tid >> 5;           // 0..7
    const int lane  = tid & 31;
    const int input = w >> 2;             // 0 = avg, 1 = max
    const int tm    = (w >> 1) & 1;
    const int tn    = w & 1;
    const float* P  = input ? maxp : avgp;

    const int row = lane & 15;            // A row / B col / D col
    const int kq  = (lane >> 4) << 1;     // K sub-offset: 0 or 2
    const int rhi = (lane >> 4) << 3;     // D row offset: 0 or 8
    const int mb  = tm * 16, nb = tn * 16;

    v8f acc = {0.f, 0.f, 0.f, 0.f, 0.f, 0.f, 0.f, 0.f};
    for (int k0 = 0; k0 < C_; k0 += 4) {
        v2f a, b;
        a.x = P[(mb + row) * C_ + k0 + kq];
        a.y = P[(mb + row) * C_ + k0 + kq + 1];
        b.x = W1[(k0 + kq) * CR_ + nb + row];
        b.y = W1[(k0 + kq + 1) * CR_ + nb + row];
        acc = __builtin_amdgcn_wmma_f32_16x16x4_f32(
            /*neg_a=*/false, a, /*neg_b=*/false, b,
            /*c_mod=*/(short)0, acc, /*reuse_a=*/false, /*reuse_b=*/false);
    }
    const float bias = b1[nb + row];
    #pragma unroll
    for (int g = 0; g < 8; ++g) {
        float v = acc[g] + bias;
        Hld[input][mb + g + rhi][nb + row] = v > 0.0f ? v : 0.0f;   // ReLU
    }
    __syncthreads();
    // HS = H_avg + H_max (second-layer linearity: (h0+h1)@W2 = h0@W2 + h1@W2)
    for (int t = tid; t < 32 * 32; t += blockDim.x)
        hs_out[t] = Hld[0][t >> 5][t & 31] + Hld[1][t >> 5][t & 31];
}

// ---------------------------------------------------------------------------
// Kernel 3: GEMM2 + sigmoid via v_wmma_f32_16x16x4_f32.
// attn = sigmoid(HS@W2 + 2*b2) -> [32][2048]. 256 tiles, 1 tile/wave.
// ---------------------------------------------------------------------------
__global__ void gemm2_kernel(const float* __restrict__ hs,
                             const float* __restrict__ W2,
                             const float* __restrict__ b2,
                             float* __restrict__ attn) {
    const int wid  = blockIdx.x * (blockDim.x >> 5) + (threadIdx.x >> 5); // 0..255
    const int lane = threadIdx.x & 31;
    const int tm   = wid & 1;
    const int tn   = wid >> 1;            // 0..127
    const int row  = lane & 15;
    const int kq   = (lane >> 4) << 1;
    const int rhi  = (lane >> 4) << 3;
    const int mb   = tm * 16, nb = tn * 16;

    v8f acc = {0.f, 0.f, 0.f, 0.f, 0.f, 0.f, 0.f, 0.f};
    #pragma unroll
    for (int k0 = 0; k0 < CR_; k0 += 4) {
        v2f a, b;
        a.x = hs[(mb + row) * CR_ + k0 + kq];
        a.y = hs[(mb + row) * CR_ + k0 + kq + 1];
        b.x = W2[(k0 + kq) * CD_ + nb + row];
        b.y = W2[(k0 + kq + 1) * CD_ + nb + row];
        acc = __builtin_amdgcn_wmma_f32_16x16x4_f32(
            false, a, false, b, (short)0, acc, false, false);
    }
    const float bias2 = 2.0f * b2[nb + row];    // b2 appears in both MLP branches
    #pragma unroll
    for (int g = 0; g < 8; ++g) {
        float v = acc[g] + bias2;
        attn[(mb + g + rhi) * CD_ + nb + row] = 1.0f / (1.0f + __expf(-v));
    }
}

// ---------------------------------------------------------------------------
// Kernel 4: stable top-128-of-512 per (n,scale) via rank counting.
// rank(c) = #{j : v[j]>v[c]} + #{j<c : v[j]==v[c]}  == lax.top_k position.
// One block (512 threads) per (n,d); values staged in LDS.
// ---------------------------------------------------------------------------
__global__ void topk_kernel(const float* __restrict__ attn,
                            int* __restrict__ idx) {
    __shared__ float vals[C_];
    const int nd = blockIdx.x;            // 0..127
    const int n  = nd >> 2;
    const int d  = nd & 3;
    const int c  = threadIdx.x;           // 0..511
    // attn[n][c*D + d]  (reference reshape(N,C,D).transpose(0,2,1))
    const float v = attn[n * CD_ + c * D_ + d];
    vals[c] = v;
    __syncthreads();
    int rank = 0;
    for (int j = 0; j < C_; ++j) {
        const float vj = vals[j];
        rank += (vj > v) || (vj == v && j < c);
    }
    if (rank < BLK_)
        idx[n * C_ + d * BLK_ + rank] = c;
}

// ---------------------------------------------------------------------------
// Kernel 5: gather selected channel planes. One block per (n,j): copy 16KB.
// Non-temporal stores keep the 256MB output stream out of L2 so x stays
// resident for cross-scale index reuse.
// ---------------------------------------------------------------------------
__global__ void gather_kernel(const float* __restrict__ x,
                              const int* __restrict__ idx,
                              float* __restrict__ out) {
    const int b  = blockIdx.x;            // n*512 + j
    const int n  = b >> 9;
    const int ch = idx[b];                // idx is laid out [n][j]
    const v4f* src = (const v4f*)(x + ((size_t)n * C_ + ch) * HW_);
    v4f*       dst = (v4f*)(out + (size_t)b * HW_);
    const int t = threadIdx.x;            // 256 threads * 4 * float4 = 4096 floats
    #pragma unroll
    for (int i = 0; i < 4; ++i) {
        v4f v = src[i * 256 + t];
        __builtin_nontemporal_store(v, &dst[i * 256 + t]);
    }
}

// ---------------------------------------------------------------------------
// Launch
// ---------------------------------------------------------------------------
extern "C" void kernel_launch(void* const* d_in, const int* in_sizes, int n_in,
                              void* d_out, int out_size, void* d_ws, size_t ws_size,
                              hipStream_t stream) {
    const float* x  = (const float*)d_in[0];   // [32,512,64,64]
    const float* W1 = (const float*)d_in[1];   // [512,32]
    const float* b1 = (const float*)d_in[2];   // [32]
    const float* W2 = (const float*)d_in[3];   // [32,2048]
    const float* b2 = (const float*)d_in[4];   // [2048]
    float* out = (float*)d_out;                // [32,512,64,64]

    // Workspace carve-up (~463 KB)
    float* avgp = (float*)d_ws;                      // 16384
    float* maxp = avgp + N_ * C_;                    // 16384
    float* hs   = maxp + N_ * C_;                    // 1024
    float* attn = hs + N_ * CR_;                     // 65536
    int*   idx  = (int*)(attn + N_ * CD_);           // 16384

    pool_kernel  <<<(N_ * C_) / 8, 256, 0, stream>>>(x, avgp, maxp);
    gemm1_kernel <<<1,             256, 0, stream>>>(avgp, maxp, W1, b1, hs);
    gemm2_kernel <<<32,            256, 0, stream>>>(hs, W2, b2, attn);
    topk_kernel  <<<N_ * D_,       C_,  0, stream>>>(attn, idx);
    gather_kernel<<<N_ * C_,       256, 0, stream>>>(x, idx, out);
}